// CRPS_53455162966139
// MI455X (gfx1250) — compile-verified
//
#include <hip/hip_runtime.h>

typedef float v2f __attribute__((ext_vector_type(2)));
typedef float v4f __attribute__((ext_vector_type(4)));
typedef float v8f __attribute__((ext_vector_type(8)));

#define NE  16      // ensemble
#define NT  8       // timesteps
#define NSP 32768   // 128*256 spatial
#define NB  2       // batch

__device__ __forceinline__ v4f vabs4(v4f v) {
    v4f r;
    r.x = __builtin_fabsf(v.x);
    r.y = __builtin_fabsf(v.y);
    r.z = __builtin_fabsf(v.z);
    r.w = __builtin_fabsf(v.w);
    return r;
}

// One thread handles 4 consecutive spatial points of one (b,t) slice.
// Total work-items: 2*8*32768/4 = 131072 -> 512 blocks x 256 threads.
__global__ __launch_bounds__(256) void crps_partial(const float* __restrict__ preds,
                                                    const float* __restrict__ target,
                                                    float* __restrict__ partial) {
    const int w    = blockIdx.x * 256 + threadIdx.x;   // 0..131071
    const int quad = w & (NSP / 4 - 1);                // 0..8191
    const int bt   = w >> 13;                          // 0..15 == b*8 + t (uniform per block)
    const int t    = bt & (NT - 1);
    const int s    = quad * 4;

    const float* p = preds + (size_t)bt * (NE * NSP) + s;
    const v4f tgt  = *(const v4f*)(target + (size_t)bt * NSP + s);

    // 16 coalesced b128 loads; ensemble stride 128KB fits signed 24-bit IOFFSET.
    v4f x[NE];
#pragma unroll
    for (int e = 0; e < NE; ++e) x[e] = *(const v4f*)(p + e * NSP);

    // term1: sum_e |x_e - tgt|
    v4f a1 = {};
#pragma unroll
    for (int e = 0; e < NE; ++e) a1 += vabs4(x[e] - tgt);

    // term2: sum_{i<j} |x_i - x_j|  (120 pairs, symmetry exploited)
    v4f a2 = {};
#pragma unroll
    for (int i = 0; i < NE; ++i)
#pragma unroll
        for (int j = i + 1; j < NE; ++j)
            a2 += vabs4(x[i] - x[j]);

    // temporal: sum_e |x_t - x_{t-1}|, attributed to thread at time t (t>=1).
    // Previous-timestep re-reads hit L2 (whole preds tensor fits in 192MB).
    v4f a3 = {};
    if (t != 0) {
        const float* pp = p - NE * NSP;
#pragma unroll
        for (int e = 0; e < NE; ++e) a3 += vabs4(x[e] - *(const v4f*)(pp + e * NSP));
    }

    const float C1 = 1.0f / 8388608.0f;     // 1/(2*32768*8*16)
    const float C2 = 1.0f / 134217728.0f;   // 1/(2*32768*8*256)
    const float C3 = 0.1f / 7340032.0f;     // 0.1/(2*32768*7*16)

    const float s1 = (a1.x + a1.y) + (a1.z + a1.w);
    const float s2 = (a2.x + a2.y) + (a2.z + a2.w);
    const float s3 = (a3.x + a3.y) + (a3.z + a3.w);
    float val = C1 * s1 - C2 * s2 + C3 * s3;

    // ---- wave32 reduction via V_WMMA_F32_16X16X4_F32 ----
    // A(16x4): lane m -> A[m,0]=val, lane m+16 -> A[m,2]=val, K=1,3 are zero.
    // B(4x16) = ones  =>  D[m,n] = val_m + val_{m+16}  for all n.
    // EXEC is all-1s here (branch above has re-converged; full 256-thread block).
    v2f a; a.x = val;  a.y = 0.0f;
    v2f b; b.x = 1.0f; b.y = 1.0f;
    v8f c = {};
    v8f d = __builtin_amdgcn_wmma_f32_16x16x4_f32(false, a, false, b,
                                                  (short)0, c, false, false);
    // Lane n (<16) holds D[0..7,n]; lane n+16 holds D[8..15,n].
    float wsum = ((d[0] + d[1]) + (d[2] + d[3])) + ((d[4] + d[5]) + (d[6] + d[7]));
    wsum += __shfl_xor(wsum, 16);            // full 32-lane sum in every lane

    __shared__ float lds[8];
    const int lane = threadIdx.x & 31;
    const int wave = threadIdx.x >> 5;
    if (lane == 0) lds[wave] = wsum;
    __syncthreads();
    if (threadIdx.x == 0) {
        float bs = 0.0f;
#pragma unroll
        for (int i = 0; i < 8; ++i) bs += lds[i];
        partial[blockIdx.x] = bs;
    }
}

// Deterministic fixed-order final reduction of the 512 per-block partials.
__global__ __launch_bounds__(256) void crps_final(const float* __restrict__ partial,
                                                  float* __restrict__ out, int n) {
    float acc = 0.0f;
    for (int i = threadIdx.x; i < n; i += 256) acc += partial[i];
#pragma unroll
    for (int off = 16; off >= 1; off >>= 1) acc += __shfl_xor(acc, off);
    __shared__ float lds[8];
    if ((threadIdx.x & 31) == 0) lds[threadIdx.x >> 5] = acc;
    __syncthreads();
    if (threadIdx.x == 0) {
        float r = 0.0f;
#pragma unroll
        for (int i = 0; i < 8; ++i) r += lds[i];
        out[0] = r;
    }
}

extern "C" void kernel_launch(void* const* d_in, const int* in_sizes, int n_in,
                              void* d_out, int out_size, void* d_ws, size_t ws_size,
                              hipStream_t stream) {
    (void)in_sizes; (void)n_in; (void)out_size; (void)ws_size;
    const float* preds  = (const float*)d_in[0];   // (2,8,16,128,256) f32
    const float* target = (const float*)d_in[1];   // (2,8,1,128,256) f32
    float* partial = (float*)d_ws;                 // 512 floats scratch
    crps_partial<<<512, 256, 0, stream>>>(preds, target, partial);
    crps_final<<<1, 256, 0, stream>>>(partial, (float*)d_out, 512);
}